// KalmanPredictorMambaRecursiveV1c_80977313399545
// MI455X (gfx1250) — compile-verified
//
#include <hip/hip_runtime.h>
#include <hip/hip_bf16.h>
#include <math.h>

typedef __attribute__((ext_vector_type(2))) float v2f;
typedef __attribute__((ext_vector_type(8))) float v8f;

#define kB  2
#define kC  64
#define kD  128
#define kH  40
#define kW  40
#define kL  (kH * kW)      // 1600
#define kK  4
#define kR  4
#define kNs 16
#define kCx (kR + 2 * kNs) // 36
#define EPSF 1e-5f

__device__ __forceinline__ float lrelu01(float v) { return v > 0.f ? v : 0.01f * v; }
__device__ __forceinline__ float siluf(float v)   { return v / (1.f + expf(-v)); }

// ---------------------------------------------------------------------------
// Generic batched f32 WMMA GEMM:  Out[m,n] = sum_k A[m,k]*B[k,n] (+ bias[m])
// A: row-major MxK (lda=K). B: row-major KxN (ldb=N). Out: row-major (ldo=N).
// Batch via blockIdx.y/z with independent element strides per operand.
// One wave per 16x16 tile; K must be a multiple of 4 (64/128/4 here).
// Uses V_WMMA_F32_16X16X4_F32 — full f32 precision matrix core path.
//
// OOB handling: A-row m feeds only D-row m, B-col n feeds only D-col n, so
// out-of-range lanes are CLAMPED to a valid address (garbage lands in
// accumulator rows/cols that are never stored). This keeps the inner loop
// free of EXEC-mask predication: loads + wmma only.
// ---------------------------------------------------------------------------
__global__ __launch_bounds__(128)
void wmma_gemm_f32_kernel(const float* __restrict__ A, int sAy, int sAz,
                          const float* __restrict__ Bm, int sBy, int sBz,
                          const float* __restrict__ bias, int sBiy,
                          float* __restrict__ Out, int sOy, int sOz,
                          int M, int N, int K)
{
    A   += (size_t)blockIdx.y * sAy + (size_t)blockIdx.z * sAz;
    Bm  += (size_t)blockIdx.y * sBy + (size_t)blockIdx.z * sBz;
    Out += (size_t)blockIdx.y * sOy + (size_t)blockIdx.z * sOz;
    if (bias) bias += (size_t)blockIdx.y * sBiy;

    const int mt   = (M + 15) >> 4;
    const int nt   = (N + 15) >> 4;
    const int wave = threadIdx.x >> 5;
    const int tile = blockIdx.x * 4 + wave;
    if (tile >= mt * nt) return;           // wave-uniform: EXEC stays all-ones
    const int tm   = (tile / nt) << 4;
    const int tn   = (tile % nt) << 4;
    const int lane = threadIdx.x & 31;
    const int half = lane >> 4;
    const int l16  = lane & 15;
    const int mc   = min(tm + l16, M - 1); // clamped A row for this lane
    const int nc   = min(tn + l16, N - 1); // clamped B col for this lane

    // A-fragment: lane holds row mc, k = k0 + 2*half + {0,1}  (8B-aligned pair)
    // B-fragment: lane holds col nc, k = k0 + 2*half + {0,1}  (rows N apart)
    const float* Ap = A  + (size_t)mc * K + (half << 1);
    const float* Bp = Bm + (size_t)(half << 1) * N + nc;

    v8f acc = {};
    for (int k0 = 0; k0 < K; k0 += 4) {
        v2f av = *(const v2f*)Ap;
        v2f bv;
        bv.x = Bp[0];
        bv.y = Bp[N];
        acc = __builtin_amdgcn_wmma_f32_16x16x4_f32(false, av, false, bv,
                                                    (short)0, acc, false, false);
        Ap += 4;
        Bp += (size_t)4 * N;
    }

    // D layout: vgpr v, lane -> element (m = tm + v + 8*half, n = tn + l16)
    const int n = tn + l16;
    if (tm + 16 <= M && tn + 16 <= N) {    // full-tile fast path (wave-uniform)
        float* op = Out + (size_t)(tm + (half << 3)) * N + n;
        #pragma unroll
        for (int v = 0; v < 8; ++v) {
            float val = acc[v];
            if (bias) val += bias[tm + (half << 3) + v];
            op[(size_t)v * N] = val;
        }
    } else {
        #pragma unroll
        for (int v = 0; v < 8; ++v) {
            const int mm = tm + (half << 3) + v;
            if (mm < M && n < N) {
                float val = acc[v];
                if (bias) val += bias[mm];
                Out[(size_t)mm * N + n] = val;
            }
        }
    }
}

// ---------------------------------------------------------------------------
// Channel LayerNorm over axis=1 of (B, Cc, L) + optional residual / leaky-relu
// / SiLU gate.  out = maybe_gate( maybe_lrelu( maybe_res + LN(x)*w + b ) )
// ---------------------------------------------------------------------------
__global__ void ln_channel_kernel(const float* __restrict__ x,
                                  const float* __restrict__ w,
                                  const float* __restrict__ b,
                                  const float* __restrict__ res,
                                  const float* __restrict__ gate, int gstrideB,
                                  float* __restrict__ out,
                                  int Bn, int Cc, int L, int do_lrelu)
{
    int idx = blockIdx.x * blockDim.x + threadIdx.x;
    if (idx >= Bn * L) return;
    int bb = idx / L, l = idx % L;
    const float* xp0 = x + (size_t)bb * Cc * L + l;
    float mu = 0.f;
    for (int c = 0; c < Cc; ++c) mu += xp0[(size_t)c * L];
    mu /= (float)Cc;
    float var = 0.f;
    for (int c = 0; c < Cc; ++c) { float d = xp0[(size_t)c * L] - mu; var += d * d; }
    var /= (float)Cc;
    float rs = rsqrtf(var + EPSF);
    for (int c = 0; c < Cc; ++c) {
        float v = (xp0[(size_t)c * L] - mu) * rs * w[c] + b[c];
        size_t o = (size_t)bb * Cc * L + (size_t)c * L + l;
        if (res)      v += res[o];
        if (do_lrelu) v  = lrelu01(v);
        if (gate)     v *= siluf(gate[(size_t)bb * gstrideB + (size_t)c * L + l]);
        out[o] = v;
    }
}

// Dense 3x3 conv, Cin=Cout=kC, SAME padding, NCHW.
__global__ void conv3x3_kernel(const float* __restrict__ x, const float* __restrict__ w,
                               const float* __restrict__ bias, float* __restrict__ out)
{
    int idx = blockIdx.x * blockDim.x + threadIdx.x;
    if (idx >= kB * kC * kL) return;
    int l = idx % kL; int oc = (idx / kL) % kC; int bb = idx / (kC * kL);
    int h = l / kW, wc = l % kW;
    float acc = bias[oc];
    const float* xb = x + (size_t)bb * kC * kL;
    const float* wb = w + (size_t)oc * kC * 9;
    for (int ic = 0; ic < kC; ++ic) {
        const float* xc = xb + (size_t)ic * kL;
        const float* wk = wb + ic * 9;
        #pragma unroll
        for (int dy = 0; dy < 3; ++dy) {
            int hh = h + dy - 1;
            if (hh < 0 || hh >= kH) continue;
            #pragma unroll
            for (int dx = 0; dx < 3; ++dx) {
                int ww = wc + dx - 1;
                if (ww < 0 || ww >= kW) continue;
                acc += wk[dy * 3 + dx] * xc[hh * kW + ww];
            }
        }
    }
    out[idx] = acc;
}

// Depthwise 3x3 conv on first kD channels of xz (B, 2kD, L) + bias + SiLU.
__global__ void dwconv_silu_kernel(const float* __restrict__ xz, const float* __restrict__ w,
                                   const float* __restrict__ b, float* __restrict__ xp)
{
    int idx = blockIdx.x * blockDim.x + threadIdx.x;
    if (idx >= kB * kD * kL) return;
    int l = idx % kL; int d = (idx / kL) % kD; int bb = idx / (kD * kL);
    int h = l / kW, wc = l % kW;
    const float* xc = xz + (size_t)bb * 2 * kD * kL + (size_t)d * kL;
    const float* wd = w + d * 9;
    float acc = b[d];
    #pragma unroll
    for (int dy = 0; dy < 3; ++dy) {
        int hh = h + dy - 1;
        if (hh < 0 || hh >= kH) continue;
        #pragma unroll
        for (int dx = 0; dx < 3; ++dx) {
            int ww = wc + dx - 1;
            if (ww < 0 || ww >= kW) continue;
            acc += wd[dy * 3 + dx] * xc[hh * kW + ww];
        }
    }
    xp[idx] = siluf(acc);
}

// xs (B,K,D,L): k=0 row-major, k=1 transposed flatten, k=2/3 reversed variants.
__global__ void build_xs_kernel(const float* __restrict__ xp, float* __restrict__ xs)
{
    int idx = blockIdx.x * blockDim.x + threadIdx.x;
    if (idx >= kB * kK * kD * kL) return;
    int l = idx % kL; int d = (idx / kL) % kD;
    int k = (idx / (kD * kL)) % kK; int bb = idx / (kK * kD * kL);
    int src;
    if (k == 0)      src = l;
    else if (k == 1) src = (l % kH) * kW + (l / kH);
    else if (k == 2) src = kL - 1 - l;
    else { int lr = kL - 1 - l; src = (lr % kH) * kW + (lr / kH); }
    xs[idx] = xp[(size_t)bb * kD * kL + (size_t)d * kL + src];
}

// kld = exp(logp)*(logp - logq), log_softmax over channel axis.
__global__ void kl_kernel(const float* __restrict__ cur, const float* __restrict__ prev,
                          float* __restrict__ out)
{
    int idx = blockIdx.x * blockDim.x + threadIdx.x;
    if (idx >= kB * kL) return;
    int bb = idx / kL, l = idx % kL;
    const float* cq = cur  + (size_t)bb * kC * kL + l;
    const float* cp = prev + (size_t)bb * kC * kL + l;
    float mq = -1e30f, mp = -1e30f;
    for (int c = 0; c < kC; ++c) {
        mq = fmaxf(mq, cq[(size_t)c * kL]);
        mp = fmaxf(mp, cp[(size_t)c * kL]);
    }
    float sq = 0.f, sp = 0.f;
    for (int c = 0; c < kC; ++c) {
        sq += expf(cq[(size_t)c * kL] - mq);
        sp += expf(cp[(size_t)c * kL] - mp);
    }
    float lq = mq + logf(sq), lp = mp + logf(sp);
    for (int c = 0; c < kC; ++c) {
        float logp = cp[(size_t)c * kL] - lp;
        float logq = cq[(size_t)c * kL] - lq;
        out[(size_t)bb * kC * kL + (size_t)c * kL + l] = expf(logp) * (logp - logq);
    }
}

__global__ void softplus_kernel(float* __restrict__ x, int n)
{
    int i = blockIdx.x * blockDim.x + threadIdx.x;
    if (i >= n) return;
    float v = x[i];
    x[i] = (v > 20.f) ? v : log1pf(expf(v));
}

// Selective-scan: 16 lanes per (b,k,d) sequence (lane = state index n).
// h[l] = exp(dt*A)*h[l-1] + dt*x*B[l];  y[l] = sum_n h*C[l,n] + Ds*x.
__global__ void scan_kernel(const float* __restrict__ delta, const float* __restrict__ xs,
                            const float* __restrict__ xdbl, const float* __restrict__ A_log,
                            const float* __restrict__ Ds, float* __restrict__ y)
{
    int t = blockIdx.x * blockDim.x + threadIdx.x;
    int gid = t >> 4;
    int n   = t & 15;
    if (gid >= kB * kK * kD) return;
    int d  = gid % kD;
    int k  = (gid / kD) % kK;
    int bb = gid / (kK * kD);
    float Av  = -expf(A_log[((size_t)k * kD + d) * kNs + n]);
    float Dsv = Ds[k * kD + d];
    size_t base = ((size_t)(bb * kK + k) * kD + d) * kL;
    const float* dl = delta + base;
    const float* xl = xs + base;
    const float* Bl = xdbl + ((size_t)(bb * kK + k) * kCx + kR + n) * kL;
    const float* Cl = xdbl + ((size_t)(bb * kK + k) * kCx + kR + kNs + n) * kL;
    float* yl = y + base;
    float h = 0.f;
    for (int l = 0; l < kL; ++l) {
        float dt = dl[l], xv = xl[l];
        float a = expf(dt * Av);
        h = a * h + dt * xv * Bl[l];
        float cb = h * Cl[l];
        cb += __shfl_xor(cb, 8, 32);   // reduce within 16-lane group (wave32)
        cb += __shfl_xor(cb, 4, 32);
        cb += __shfl_xor(cb, 2, 32);
        cb += __shfl_xor(cb, 1, 32);
        if (n == 0) yl[l] = cb + Dsv * xv;
    }
}

// ysum[b,d,l] = y[b,0,d,l] + y[b,2,d,L-1-l] + untrans(y[b,1]) + untrans(y[b,3,rev])
__global__ void ysum_kernel(const float* __restrict__ y, float* __restrict__ ys)
{
    int idx = blockIdx.x * blockDim.x + threadIdx.x;
    if (idx >= kB * kD * kL) return;
    int l = idx % kL; int d = (idx / kL) % kD; int bb = idx / (kD * kL);
    int tr = (l % kW) * kH + (l / kW);
    const size_t kstride = (size_t)kD * kL;
    size_t base0 = (((size_t)bb * kK) * kD + d) * kL;
    float v = y[base0 + 0 * kstride + l]
            + y[base0 + 2 * kstride + (kL - 1 - l)]
            + y[base0 + 1 * kstride + tr]
            + y[base0 + 3 * kstride + (kL - 1 - tr)];
    ys[idx] = v;
}

__global__ void update_kernel(const float* __restrict__ cur, const float* __restrict__ sg,
                              const float* __restrict__ bs, float* __restrict__ nxt)
{
    int i = blockIdx.x * blockDim.x + threadIdx.x;
    if (i >= kB * kC * kL) return;
    nxt[i] = cur[i] * expf(sg[i]) + bs[i];
}

__global__ void sin_kernel(const float* __restrict__ x, float* __restrict__ out)
{
    int i = blockIdx.x * blockDim.x + threadIdx.x;
    if (i >= kB * kC * kL) return;
    out[i] = sinf(x[i]);
}

// ---------------------------------------------------------------------------
// ss2d host orchestration
// ---------------------------------------------------------------------------
static void ss2d_run(const float* xin, const float* const* prm, float* out,
                     float* xz, float* xp, float* xs, float* xdbl, float* delta,
                     float* ybuf, float* ysum, float* ytn, hipStream_t stream)
{
    const float* in_proj_w = prm[0];
    const float* in_proj_b = prm[1];
    const float* conv_w    = prm[2];
    const float* conv_b    = prm[3];
    const float* x_proj_w  = prm[4];
    const float* dt_w      = prm[5];
    const float* dt_b      = prm[6];
    const float* A_log     = prm[7];
    const float* Ds        = prm[8];
    const float* onw       = prm[9];
    const float* onb       = prm[10];
    const float* out_proj_w = prm[11];
    const float* out_proj_b = prm[12];
    const int TP = 256;

    // 1) in_proj: (2D x C) @ (C x L) -> xz (B, 2D, L)       [WMMA]
    {
        dim3 g(400, 1, kB);   // 16*100 tiles / 4 waves
        wmma_gemm_f32_kernel<<<g, 128, 0, stream>>>(
            in_proj_w, 0, 0, xin, 0, kC * kL, in_proj_b, 0,
            xz, 0, 2 * kD * kL, 2 * kD, kL, kC);
    }
    // 2) depthwise conv + silu -> xp
    dwconv_silu_kernel<<<(kB * kD * kL + TP - 1) / TP, TP, 0, stream>>>(xz, conv_w, conv_b, xp);
    // 3) 4-direction scan inputs
    build_xs_kernel<<<(kB * kK * kD * kL + TP - 1) / TP, TP, 0, stream>>>(xp, xs);
    // 4) x_proj: (36 x 128) @ (128 x L) per (b,k) -> xdbl    [WMMA]
    {
        dim3 g(75, kK, kB);   // 3*100 tiles / 4 waves
        wmma_gemm_f32_kernel<<<g, 128, 0, stream>>>(
            x_proj_w, kCx * kD, 0, xs, kD * kL, kK * kD * kL, nullptr, 0,
            xdbl, kCx * kL, kK * kCx * kL, kCx, kL, kD);
    }
    // 5) dt-proj: (128 x 4) @ (4 x L) + dt_b -> delta        [WMMA]
    {
        dim3 g(200, kK, kB);  // 8*100 tiles / 4 waves
        wmma_gemm_f32_kernel<<<g, 128, 0, stream>>>(
            dt_w, kD * kR, 0, xdbl, kCx * kL, kK * kCx * kL, dt_b, kD,
            delta, kD * kL, kK * kD * kL, kD, kL, kR);
    }
    // 6) softplus in place
    softplus_kernel<<<(kB * kK * kD * kL + TP - 1) / TP, TP, 0, stream>>>(delta, kB * kK * kD * kL);
    // 7) selective scan -> ybuf
    scan_kernel<<<(kB * kK * kD * 16 + TP - 1) / TP, TP, 0, stream>>>(delta, xs, xdbl, A_log, Ds, ybuf);
    // 8) combine 4 directions -> ysum (B, D, L)
    ysum_kernel<<<(kB * kD * kL + TP - 1) / TP, TP, 0, stream>>>(ybuf, ysum);
    // 9) out_norm (LN over D) * silu(z) -> ytn ; z = xz[:, D:]
    ln_channel_kernel<<<(kB * kL + TP - 1) / TP, TP, 0, stream>>>(
        ysum, onw, onb, nullptr, xz + (size_t)kD * kL, 2 * kD * kL, ytn, kB, kD, kL, 0);
    // 10) out_proj: (64 x 128) @ (128 x L) + b -> out (B,C,L) [WMMA]
    {
        dim3 g(100, 1, kB);   // 4*100 tiles / 4 waves
        wmma_gemm_f32_kernel<<<g, 128, 0, stream>>>(
            out_proj_w, 0, 0, ytn, 0, kD * kL, out_proj_b, 0,
            out, 0, kC * kL, kC, kL, kD);
    }
}

extern "C" void kernel_launch(void* const* d_in, const int* in_sizes, int n_in,
                              void* d_out, int out_size, void* d_ws, size_t ws_size,
                              hipStream_t stream)
{
    (void)in_sizes; (void)n_in; (void)out_size; (void)ws_size;
    // setup_inputs() leaf order: image, convp{w1,b1,ln1w,ln1b,w2,b2,ln2w,ln2b},
    // nf_w, nf_b, msp{13 leaves}, mbp{13 leaves}
    const float* image = (const float*)d_in[0];
    const float* w1   = (const float*)d_in[1];
    const float* b1   = (const float*)d_in[2];
    const float* ln1w = (const float*)d_in[3];
    const float* ln1b = (const float*)d_in[4];
    const float* w2   = (const float*)d_in[5];
    const float* b2   = (const float*)d_in[6];
    const float* ln2w = (const float*)d_in[7];
    const float* ln2b = (const float*)d_in[8];
    const float* nf_w = (const float*)d_in[9];
    const float* nf_b = (const float*)d_in[10];
    const float* msp[13]; for (int i = 0; i < 13; ++i) msp[i] = (const float*)d_in[11 + i];
    const float* mbp[13]; for (int i = 0; i < 13; ++i) mbp[i] = (const float*)d_in[24 + i];

    // carve workspace (all f32)
    float* wsp = (float*)d_ws;
    size_t off = 0;
    auto alloc = [&](size_t n) { float* p = wsp + off; off += n; return p; };
    const size_t nImg = (size_t)kB * kC * kL;
    float* cur   = alloc(nImg);
    float* prev  = alloc(nImg);
    float* nxt   = alloc(nImg);
    float* tmp1  = alloc(nImg);
    float* tmp2  = alloc(nImg);
    float* kld   = alloc(nImg);
    float* sigma = alloc(nImg);
    float* biasb = alloc(nImg);
    float* xz    = alloc((size_t)kB * 2 * kD * kL);
    float* xp    = alloc((size_t)kB * kD * kL);
    float* xs    = alloc((size_t)kB * kK * kD * kL);
    float* xdbl  = alloc((size_t)kB * kK * kCx * kL);
    float* delta = alloc((size_t)kB * kK * kD * kL);
    float* ybuf  = alloc((size_t)kB * kK * kD * kL);
    float* ysum  = alloc((size_t)kB * kD * kL);
    float* ytn   = alloc((size_t)kB * kD * kL);

    const int TP = 256;
    // current = layernorm2d(image, nf_w, nf_b)
    ln_channel_kernel<<<(kB * kL + TP - 1) / TP, TP, 0, stream>>>(
        image, nf_w, nf_b, nullptr, nullptr, 0, cur, kB, kC, kL, 0);
    // previous = resblock(image)
    conv3x3_kernel<<<(nImg + TP - 1) / TP, TP, 0, stream>>>(image, w1, b1, tmp1);
    ln_channel_kernel<<<(kB * kL + TP - 1) / TP, TP, 0, stream>>>(
        tmp1, ln1w, ln1b, nullptr, nullptr, 0, tmp2, kB, kC, kL, 1);
    conv3x3_kernel<<<(nImg + TP - 1) / TP, TP, 0, stream>>>(tmp2, w2, b2, tmp1);
    ln_channel_kernel<<<(kB * kL + TP - 1) / TP, TP, 0, stream>>>(
        tmp1, ln2w, ln2b, image, nullptr, 0, prev, kB, kC, kL, 1);

    for (int it = 0; it < 2; ++it) {
        kl_kernel<<<(kB * kL + TP - 1) / TP, TP, 0, stream>>>(cur, prev, kld);
        ss2d_run(kld, msp, sigma, xz, xp, xs, xdbl, delta, ybuf, ysum, ytn, stream);
        ss2d_run(cur, mbp, biasb, xz, xp, xs, xdbl, delta, ybuf, ysum, ytn, stream);
        update_kernel<<<(nImg + TP - 1) / TP, TP, 0, stream>>>(cur, sigma, biasb, nxt);
        float* t = prev; prev = cur; cur = nxt; nxt = t;   // rotate buffers
    }
    sin_kernel<<<(nImg + TP - 1) / TP, TP, 0, stream>>>(cur, (float*)d_out);
}